// WavetableSynth_29686813950361
// MI455X (gfx1250) — compile-verified
//
#include <hip/hip_runtime.h>
#include <hip/hip_bf16.h>

// ---------------- CDNA5 wavetable synth ----------------
// out[t] = amplitude[t] * lerp(meanWT, phase[t]),  phase = excl-cumsum((L/SR)*pitch) mod 512
// 3-pass scan, block offsets carried mod 512 in double, meanWT via v_wmma_f32_16x16x4_f32,
// LDS staging via global_load_async_to_lds_b32 (ASYNCcnt path). wave32 throughout.

typedef __attribute__((ext_vector_type(2))) float v2f;
typedef __attribute__((ext_vector_type(8))) float v8f;

#define BLOCK   256
#define ITEMS   16
#define CHUNK   (BLOCK * ITEMS)       // 4096 samples per block
#define PHK     (512.0f / 44100.0f)   // L / SR

// ---------- pass 1: per-block sum of inc ----------
__global__ void k_blocksum(const float* __restrict__ pitch, float* __restrict__ bsum, int T) {
  const int tid = threadIdx.x;
  const int base = blockIdx.x * CHUNK + tid * ITEMS;
  float s = 0.0f;
  if (base + ITEMS <= T) {
    const float4* p4 = (const float4*)(pitch + base);   // 16B aligned (base % 16 == 0)
#pragma unroll
    for (int i = 0; i < ITEMS / 4; ++i) { float4 v = p4[i]; s += v.x + v.y + v.z + v.w; }
  } else {
    for (int i = 0; i < ITEMS; ++i) { int t = base + i; if (t < T) s += pitch[t]; }
  }
  s *= PHK;
  __shared__ float red[BLOCK];
  red[tid] = s; __syncthreads();
  for (int off = BLOCK / 2; off > 0; off >>= 1) {
    if (tid < off) red[tid] += red[tid + off];
    __syncthreads();
  }
  if (tid == 0) bsum[blockIdx.x] = red[0];
}

// ---------- pass 2: WMMA column-mean of wavetables + double scan of block sums ----------
__global__ void k_scan_meanwt(const float* __restrict__ bsum, const float* __restrict__ wt,
                              float* __restrict__ boff, float* __restrict__ meanwt, int NB) {
  const int tid  = threadIdx.x;          // 512 threads = 16 full waves (EXEC all-ones for WMMA)
  const int wave = tid >> 5;
  const int lane = tid & 31;
  const int half = lane >> 4;            // 0: lanes 0-15, 1: lanes 16-31
  const int n    = lane & 15;

  // --- meanWT[j] = (1/64) * sum_k WT[k][j] via D = A(16x4 of 1/64) x B(4x16 of WT) + C ---
  // A is uniform, so only the column (N=lane&15) mapping of B matters; each of the 64 rows
  // is fed into exactly one K-slot across 16 accumulation steps.
  v2f a;  a[0] = 1.0f / 64.0f; a[1] = 1.0f / 64.0f;
  v8f c0 = {}; v8f c1 = {};
  const int colA = wave * 32;
  const int colB = colA + 16;
#pragma unroll
  for (int i = 0; i < 16; ++i) {
    const int r = 4 * i + 2 * half;      // rows r, r+1 for this half-wave
    v2f bA; bA[0] = wt[r * 512 + colA + n]; bA[1] = wt[(r + 1) * 512 + colA + n];
    v2f bB; bB[0] = wt[r * 512 + colB + n]; bB[1] = wt[(r + 1) * 512 + colB + n];
    c0 = __builtin_amdgcn_wmma_f32_16x16x4_f32(false, a, false, bA, (short)0, c0, false, false);
    c1 = __builtin_amdgcn_wmma_f32_16x16x4_f32(false, a, false, bB, (short)0, c1, false, false);
  }
  // D layout: VGPR0, lanes 0-15 = (M=0, N=lane) -> meanWT[col + lane]
  if (half == 0) { meanwt[colA + n] = c0[0]; meanwt[colB + n] = c1[0]; }

  // --- exclusive scan of block sums, double precision, offsets stored mod 512 ---
  __shared__ double sc[512];
  __shared__ double carry_s;
  if (tid == 0) carry_s = 0.0;
  __syncthreads();
  for (int basei = 0; basei < NB; basei += 512) {
    const int i = basei + tid;
    const double v = (i < NB) ? (double)bsum[i] : 0.0;
    sc[tid] = v; __syncthreads();
    for (int off = 1; off < 512; off <<= 1) {
      const double add = (tid >= off) ? sc[tid - off] : 0.0;
      __syncthreads();
      sc[tid] += add;
      __syncthreads();
    }
    const double carry = carry_s;
    if (i < NB) boff[i] = (float)fmod(carry + (sc[tid] - v), 512.0);  // exclusive, mod L
    __syncthreads();
    if (tid == 511) carry_s = carry + sc[511];
    __syncthreads();
  }
}

// ---------- pass 3: intra-block scan + gather-interp synth ----------
__device__ __forceinline__ float synth_one(float& ph, float incr, float amp,
                                           const float* __restrict__ s_wt) {
  float p = ph; ph = p + incr;                       // serial phase recurrence, pre-increment value
  p = p - floorf(p * (1.0f / 512.0f)) * 512.0f;      // fast positive fmod
  if (p >= 512.0f) p -= 512.0f;
  const int lo = (int)p;
  const float fr = p - (float)lo;
  const int hi = (lo + 1) & 511;
  return (s_wt[lo] * (1.0f - fr) + s_wt[hi] * fr) * amp;
}

__global__ void k_synth(const float* __restrict__ pitch, const float* __restrict__ amp,
                        const float* __restrict__ boff, const float* __restrict__ meanwt,
                        float* __restrict__ out, int T) {
  __shared__ float s_wt[512];
  __shared__ float s_wsum[BLOCK / 32];
  const int tid = threadIdx.x;

  // Stage meanWT -> LDS via async copy (ASYNCcnt path). Low 32 bits of a flat
  // pointer into LDS are the LDS byte address (ISA 10.2 aperture mapping).
  {
    unsigned lds0 = (unsigned)(size_t)&s_wt[tid];
    unsigned long long g0 = (unsigned long long)(size_t)(meanwt + tid);
    asm volatile("global_load_async_to_lds_b32 %0, %1, off" :: "v"(lds0), "v"(g0) : "memory");
    unsigned lds1 = (unsigned)(size_t)&s_wt[tid + 256];
    unsigned long long g1 = (unsigned long long)(size_t)(meanwt + tid + 256);
    asm volatile("global_load_async_to_lds_b32 %0, %1, off" :: "v"(lds1), "v"(g1) : "memory");
    asm volatile("s_wait_asynccnt 0" ::: "memory");
  }

  const int base = blockIdx.x * CHUNK + tid * ITEMS;
  const bool full = (base + ITEMS <= T);

  // load + scale increments
  float vals[ITEMS];
  if (full) {
    const float4* p4 = (const float4*)(pitch + base);
#pragma unroll
    for (int i = 0; i < ITEMS / 4; ++i) {
      float4 v = p4[i];
      vals[4*i+0] = v.x * PHK; vals[4*i+1] = v.y * PHK;
      vals[4*i+2] = v.z * PHK; vals[4*i+3] = v.w * PHK;
    }
  } else {
    for (int i = 0; i < ITEMS; ++i) { int t = base + i; vals[i] = (t < T) ? pitch[t] * PHK : 0.0f; }
  }
  float tsum = 0.0f;
#pragma unroll
  for (int i = 0; i < ITEMS; ++i) tsum += vals[i];

  // wave32 inclusive scan of per-thread sums
  const int ln = tid & 31, wv = tid >> 5;
  float incs = tsum;
#pragma unroll
  for (int d = 1; d < 32; d <<= 1) {
    float up = __shfl_up(incs, d, 32);
    if (ln >= d) incs += up;
  }
  if (ln == 31) s_wsum[wv] = incs;
  __syncthreads();
  if (wv == 0) {
    float w = (ln < BLOCK / 32) ? s_wsum[ln] : 0.0f;
#pragma unroll
    for (int d = 1; d < BLOCK / 32; d <<= 1) {
      float up = __shfl_up(w, d, 32);
      if (ln >= d) w += up;
    }
    if (ln < BLOCK / 32) s_wsum[ln] = w;     // inclusive wave sums
  }
  __syncthreads();
  const float wave_off = (wv > 0) ? s_wsum[wv - 1] : 0.0f;
  float phase = boff[blockIdx.x] + wave_off + (incs - tsum);  // exclusive prefix for this thread

  // synthesize + write
  if (full) {
    const float4* a4 = (const float4*)(amp + base);
    float4* o4 = (float4*)(out + base);
#pragma unroll
    for (int i = 0; i < ITEMS / 4; ++i) {
      float4 av = a4[i];
      float4 ov;
      ov.x = synth_one(phase, vals[4*i+0], av.x, s_wt);
      ov.y = synth_one(phase, vals[4*i+1], av.y, s_wt);
      ov.z = synth_one(phase, vals[4*i+2], av.z, s_wt);
      ov.w = synth_one(phase, vals[4*i+3], av.w, s_wt);
      o4[i] = ov;
    }
  } else {
    for (int i = 0; i < ITEMS; ++i) {
      int t = base + i;
      if (t < T) out[t] = synth_one(phase, vals[i], amp[t], s_wt);
      else { float dummy = phase; (void)dummy; }
    }
  }
}

extern "C" void kernel_launch(void* const* d_in, const int* in_sizes, int n_in,
                              void* d_out, int out_size, void* d_ws, size_t ws_size,
                              hipStream_t stream) {
  const float* pitch = (const float*)d_in[0];
  const float* amp   = (const float*)d_in[1];
  const float* wt    = (const float*)d_in[2];   // 64 x 512 row-major
  float* out = (float*)d_out;
  const int T  = in_sizes[0];
  const int NB = (T + CHUNK - 1) / CHUNK;

  float* ws_f   = (float*)d_ws;
  float* bsum   = ws_f;                 // NB floats
  float* boff   = ws_f + NB;            // NB floats (block offsets mod 512)
  float* meanwt = ws_f + 2 * NB;        // 512 floats

  k_blocksum   <<<NB, BLOCK, 0, stream>>>(pitch, bsum, T);
  k_scan_meanwt<<<1, 512, 0, stream>>>(bsum, wt, boff, meanwt, NB);
  k_synth      <<<NB, BLOCK, 0, stream>>>(pitch, amp, boff, meanwt, out, T);
}